// Attention_56624848831195
// MI455X (gfx1250) — compile-verified
//
#include <hip/hip_runtime.h>
#include <stdint.h>

// CDNA5 (gfx1250) fp32 WMMA: D(16x16,f32) = A(16x4,f32) x B(4x16,f32) + C
typedef __attribute__((ext_vector_type(2))) float v2f;
typedef __attribute__((ext_vector_type(8))) float v8f;

#define WMMA_F32(a, b, c) \
  __builtin_amdgcn_wmma_f32_16x16x4_f32(false, (a), false, (b), (short)0, (c), false, false)

// Async Global -> LDS copy (gfx1250), 8 bytes per lane, tracked by ASYNCcnt.
// ldsoff = workgroup-relative LDS byte offset (low 32 bits of generic pointer),
// gaddr  = 64-bit global address.
__device__ __forceinline__ void async_ld_b64(uint32_t ldsoff, uint64_t gaddr) {
  asm volatile("global_load_async_to_lds_b64 %0, %1, off"
               :: "v"(ldsoff), "v"(gaddr) : "memory");
}
__device__ __forceinline__ void wait_async0() {
  asm volatile("s_wait_asynccnt 0" ::: "memory");
}
__device__ __forceinline__ uint32_t lds_off(const void* p) {
  return (uint32_t)(size_t)p;
}

// ---------------------------------------------------------------------------
// Generic C[M,N] = X[M,K] * W[N,K]^T   (K = 1024, block tile 64x64, 4 waves)
// Double-buffered async LDS staging of 64x16 slabs of X and W.
// MODE 0: plain row-major store (N stride 1024) into `out`
// MODE 1: QKV store: column n -> (which = n/1024, head = (n%1024)/64, hd),
//         row m -> (b = m/2048, l = m%2048); writes [B,H,L,64] q/k/v buffers
// ---------------------------------------------------------------------------
template <int MODE>
__global__ __launch_bounds__(128) void gemm_xwt(
    const float* __restrict__ X, const float* __restrict__ W,
    float* __restrict__ out, float* __restrict__ qb, float* __restrict__ kb,
    float* __restrict__ vb) {
  constexpr int K = 1024;
  constexpr int LDA = 18;  // padded LDS stride (even -> b64 aligned, conflict-free)
  __shared__ __align__(16) float xs[2][64 * LDA];
  __shared__ __align__(16) float wsm[2][64 * LDA];

  const int tid = threadIdx.x;
  const int wave = tid >> 5;
  const int lane = tid & 31;
  const int lm = lane & 15;   // N (or M) index within tile
  const int lg = lane >> 4;   // half-wave select
  const int m0 = blockIdx.y * 64;
  const int n0 = blockIdx.x * 64;

  auto stage = [&](int buf, int k0) {
#pragma unroll
    for (int i = 0; i < 4; ++i) {
      int idx = tid + i * 128;  // 0..511 float2 slots per array
      int row = idx >> 3;       // 8 float2 per 16-float row
      int c2 = (idx & 7) * 2;
      async_ld_b64(lds_off(&xs[buf][row * LDA + c2]),
                   (uint64_t)(size_t)&X[(size_t)(m0 + row) * K + k0 + c2]);
      async_ld_b64(lds_off(&wsm[buf][row * LDA + c2]),
                   (uint64_t)(size_t)&W[(size_t)(n0 + row) * K + k0 + c2]);
    }
  };

  const v8f vzero = {0, 0, 0, 0, 0, 0, 0, 0};
  v8f acc[4] = {vzero, vzero, vzero, vzero};

  stage(0, 0);
  for (int k0 = 0; k0 < K; k0 += 16) {
    const int cur = (k0 >> 4) & 1;
    wait_async0();       // this wave's async writes to buf[cur] done
    __syncthreads();     // all waves' writes visible; buf[cur^1] free to reuse
    if (k0 + 16 < K) stage(cur ^ 1, k0 + 16);  // prefetch next slab
#pragma unroll
    for (int s = 0; s < 4; ++s) {  // 4 K-steps of 4 inside the 16-slab
      v2f a = *(v2f*)&xs[cur][(wave * 16 + lm) * LDA + s * 4 + lg * 2];
#pragma unroll
      for (int nt = 0; nt < 4; ++nt) {
        v2f b = *(v2f*)&wsm[cur][(nt * 16 + lm) * LDA + s * 4 + lg * 2];
        acc[nt] = WMMA_F32(a, b, acc[nt]);
      }
    }
  }

#pragma unroll
  for (int nt = 0; nt < 4; ++nt) {
#pragma unroll
    for (int r = 0; r < 8; ++r) {
      int mrow = m0 + wave * 16 + r + 8 * lg;  // C/D: VGPR r -> M = r + 8*lg
      int ncol = n0 + nt * 16 + lm;
      float v = acc[nt][r];
      if (MODE == 0) {
        out[(size_t)mrow * 1024 + ncol] = v;
      } else {
        int which = ncol >> 10;  // 0=q 1=k 2=v
        int d = ncol & 1023;
        int h = d >> 6, hd = d & 63;
        int b = mrow >> 11, l = mrow & 2047;
        float* dst = (which == 0) ? qb : ((which == 1) ? kb : vb);
        dst[(((size_t)(b * 16 + h)) * 2048 + l) * 64 + hd] = v;
      }
    }
  }
}

// ---------------------------------------------------------------------------
// Causal flash attention. Q/K/V in [B,H,L,64]. Block = 64 query rows
// (4 waves x 16-row m-tile), loops over 16-row KV tiles with online softmax.
// K/V tiles double-buffered through LDS with async Global->LDS copies.
// Output written to attn buffer [B,L,1024] (heads re-interleaved).
// ---------------------------------------------------------------------------
__global__ __launch_bounds__(128) void flash_attn(
    const float* __restrict__ qb, const float* __restrict__ kb,
    const float* __restrict__ vb, float* __restrict__ ab) {
  constexpr int L = 2048, Hd = 64;
  constexpr int LDK = 66;  // padded K/V tile stride
  constexpr int LDP = 18;  // padded P staging stride
  __shared__ __align__(16) float ks[2][16 * LDK];
  __shared__ __align__(16) float vs[2][16 * LDK];
  __shared__ __align__(16) float ps[4][16 * LDP];

  const int tid = threadIdx.x;
  const int wave = tid >> 5;
  const int lane = tid & 31;
  const int lm = lane & 15;
  const int lg = lane >> 4;
  const int bh = blockIdx.x;    // b*16 + h
  const int mblk = blockIdx.y;  // 64-row block
  const int m0 = mblk * 64 + wave * 16;
  const size_t base = (size_t)bh * L * Hd;
  const float* Q = qb + base;
  const float* Kp = kb + base;
  const float* Vp = vb + base;

  auto stage_kv = [&](int buf, int kv0) {
#pragma unroll
    for (int i = 0; i < 4; ++i) {
      int idx = tid + i * 128;  // 0..511 float2 slots per array
      int row = idx >> 5;       // 32 float2 per 64-float row
      int c2 = (idx & 31) * 2;
      async_ld_b64(lds_off(&ks[buf][row * LDK + c2]),
                   (uint64_t)(size_t)&Kp[(size_t)(kv0 + row) * Hd + c2]);
      async_ld_b64(lds_off(&vs[buf][row * LDK + c2]),
                   (uint64_t)(size_t)&Vp[(size_t)(kv0 + row) * Hd + c2]);
    }
  };

  // Preload this wave's 16x64 Q tile as A-fragments (16 k-steps of 4).
  v2f qf[16];
#pragma unroll
  for (int s = 0; s < 16; ++s)
    qf[s] = *(const v2f*)&Q[(size_t)(m0 + lm) * Hd + s * 4 + lg * 2];

  const v8f vzero = {0, 0, 0, 0, 0, 0, 0, 0};
  v8f o[4] = {vzero, vzero, vzero, vzero};
  float rmax[8], rsum[8];
#pragma unroll
  for (int r = 0; r < 8; ++r) {
    rmax[r] = -3.0e38f;
    rsum[r] = 0.0f;
  }

  const int kv_end = mblk * 64 + 64;  // block-wide causal bound
  stage_kv(0, 0);
  for (int kv0 = 0; kv0 < kv_end; kv0 += 16) {
    const int cur = (kv0 >> 4) & 1;
    wait_async0();
    __syncthreads();  // buf[cur] published; buf[cur^1] free to overwrite
    if (kv0 + 16 < kv_end) stage_kv(cur ^ 1, kv0 + 16);

    // S = Q * K^T : accumulate 16 k-steps over head dim.
    v8f sa = vzero;
#pragma unroll
    for (int s = 0; s < 16; ++s) {
      v2f b = *(v2f*)&ks[cur][lm * LDK + s * 4 + lg * 2];
      sa = WMMA_F32(qf[s], b, sa);
    }

    // Online softmax (scale 1/sqrt(64)=0.125, analytic causal mask).
#pragma unroll
    for (int r = 0; r < 8; ++r) {
      int mrow = m0 + r + 8 * lg;
      int ncol = kv0 + lm;
      float sv = sa[r] * 0.125f;
      sv = (ncol <= mrow) ? sv : -3.0e38f;
      float tm = sv;
#pragma unroll
      for (int off = 1; off < 16; off <<= 1)
        tm = fmaxf(tm, __shfl_xor(tm, off, 32));
      float nm = fmaxf(rmax[r], tm);
      float corr = __expf(rmax[r] - nm);
      float p = __expf(sv - nm);
      float psum = p;
#pragma unroll
      for (int off = 1; off < 16; off <<= 1) psum += __shfl_xor(psum, off, 32);
      rsum[r] = rsum[r] * corr + psum;
      rmax[r] = nm;
#pragma unroll
      for (int dt = 0; dt < 4; ++dt) o[dt][r] *= corr;
      // Stage P (C-layout) to LDS so it can be re-read in A-layout.
      ps[wave][(r + 8 * lg) * LDP + lm] = p;
    }
    asm volatile("s_wait_dscnt 0" ::: "memory");  // same-wave LDS RAW fence

    // O += P * V : K-dim is the 16 KV rows -> 4 k-steps, 4 d-tiles of 16.
#pragma unroll
    for (int s = 0; s < 4; ++s) {
      v2f a = *(v2f*)&ps[wave][lm * LDP + s * 4 + lg * 2];
#pragma unroll
      for (int dt = 0; dt < 4; ++dt) {
        v2f b;
        b.x = vs[cur][(s * 4 + lg * 2 + 0) * LDK + dt * 16 + lm];
        b.y = vs[cur][(s * 4 + lg * 2 + 1) * LDK + dt * 16 + lm];
        o[dt] = WMMA_F32(a, b, o[dt]);
      }
    }
  }

  // Normalize and scatter to [B, L, D] with heads interleaved.
  const int b = bh >> 4, h = bh & 15;
#pragma unroll
  for (int r = 0; r < 8; ++r) {
    float inv = 1.0f / rsum[r];
    int mrow = m0 + r + 8 * lg;
#pragma unroll
    for (int dt = 0; dt < 4; ++dt) {
      ab[((size_t)(b * L + mrow)) * 1024 + h * 64 + dt * 16 + lm] =
          o[dt][r] * inv;
    }
  }
}

// ---------------------------------------------------------------------------
extern "C" void kernel_launch(void* const* d_in, const int* in_sizes, int n_in,
                              void* d_out, int out_size, void* d_ws,
                              size_t ws_size, hipStream_t stream) {
  (void)in_sizes; (void)n_in; (void)out_size; (void)ws_size;
  const float* x = (const float*)d_in[0];      // [2,2048,1024]
  // d_in[1] = boolean causal mask: applied analytically, not read.
  const float* w_qkv = (const float*)d_in[2];  // [3072,1024]
  const float* w_out = (const float*)d_in[3];  // [1024,1024]
  float* out = (float*)d_out;                  // [2,2048,1024]

  const size_t SZ = (size_t)4 * 1024 * 1024;   // B*L*D floats
  float* qb = (float*)d_ws;                    // [B,H,L,64]
  float* kb = qb + SZ;
  float* vb = kb + SZ;
  float* ab = vb + SZ;                         // [B,L,D]

  dim3 blk(128, 1, 1);
  // qkv = x @ w_qkv^T  (M=4096, N=3072) -> split-head Q/K/V
  gemm_xwt<1><<<dim3(48, 64, 1), blk, 0, stream>>>(x, w_qkv, nullptr, qb, kb, vb);
  // attention: grid (B*H, L/64)
  flash_attn<<<dim3(32, 32, 1), blk, 0, stream>>>(qb, kb, vb, ab);
  // out = attn @ w_out^T  (M=4096, N=1024)
  gemm_xwt<0><<<dim3(16, 64, 1), blk, 0, stream>>>(ab, w_out, out, nullptr,
                                                   nullptr, nullptr);
}